// GCN_edge_19378892440059
// MI455X (gfx1250) — compile-verified
//
#include <hip/hip_runtime.h>

#define NNODES 100000
#define NEDGES 1600000
#define DD     32
#define RS     104        // LDS row stride in floats (16B aligned, bank-spreading)
#define ROWS   16         // edge rows per wave (WMMA M)
#define WAVES  8          // waves per block
#define TPB    256
#define EBLK   (WAVES*ROWS)   // 128 edges per block; 1.6M/128 = 12500 exactly

typedef __attribute__((ext_vector_type(2))) float v2f;
typedef __attribute__((ext_vector_type(8))) float v8f;

__device__ __forceinline__ v8f wmma4(v2f a, v2f b, v8f c) {
  // D = A(16x4) * B(4x16) + C(16x16), all f32.  8-arg form:
  // (neg_a, A, neg_b, B, c_mod, C, reuse_a, reuse_b)
  return __builtin_amdgcn_wmma_f32_16x16x4_f32(false, a, false, b, (short)0, c,
                                               false, false);
}

// One 16-row x (NT*16)-col GEMM over K features staged in LDS rows.
// A layout: lane(g=lane>>4, m=lane&15): K = 4*step + 2g + vgpr
// B layout: lane:                      K = 4*step + 2g + vgpr, N = lane&15 (+16*nt)
// C/D layout: m = v + 8g, n = (lane&15) + 16*nt
// Fragments are preloaded into register arrays so the WMMA chain issues
// back-to-back (single loadcnt/dscnt wait up front instead of per-step waits).
template<int K, int NW, int NT>
__device__ __forceinline__ void gemm_tile(const float* tile,
                                          const float* __restrict__ W,
                                          const float* __restrict__ bias,
                                          v8f* acc) {
  constexpr int STEPS = (K + 3) / 4;
  const int lane = threadIdx.x & 31;
  const int g    = lane >> 4;
  const int col  = lane & 15;

  v2f a[STEPS];
  v2f b[NT][STEPS];
#pragma unroll
  for (int s = 0; s < STEPS; ++s) {
    const int ka = 4 * s + 2 * g;
    a[s].x = (ka     < K) ? tile[col * RS + ka]     : 0.0f;
    a[s].y = (ka + 1 < K) ? tile[col * RS + ka + 1] : 0.0f;
#pragma unroll
    for (int nt = 0; nt < NT; ++nt) {
      const int n = nt * 16 + col;
      b[nt][s].x = (ka     < K && n < NW) ? W[ka       * NW + n] : 0.0f;
      b[nt][s].y = (ka + 1 < K && n < NW) ? W[(ka + 1) * NW + n] : 0.0f;
    }
  }

#pragma unroll
  for (int nt = 0; nt < NT; ++nt) {
    float binit = 0.0f;
    if (bias) {
      const int n = nt * 16 + col;
      binit = (n < NW) ? bias[n] : 0.0f;
    }
    v8f c;
#pragma unroll
    for (int v = 0; v < 8; ++v) c[v] = binit;
    acc[nt] = c;
  }

#pragma unroll
  for (int s = 0; s < STEPS; ++s)
#pragma unroll
    for (int nt = 0; nt < NT; ++nt)
      acc[nt] = wmma4(a[s], b[nt][s], acc[nt]);
}

template<int NT>
__device__ __forceinline__ void tile_to_lds(float* tile, const v8f* acc, bool dorelu) {
  const int lane = threadIdx.x & 31;
  const int g    = lane >> 4;
  const int col  = lane & 15;
#pragma unroll
  for (int nt = 0; nt < NT; ++nt)
#pragma unroll
    for (int v = 0; v < 8; ++v) {
      float x = acc[nt][v];
      if (dorelu) x = fmaxf(x, 0.0f);
      tile[(v + 8 * g) * RS + nt * 16 + col] = x;
    }
}

// ---------------- node encoder: tiny (0.3 GFLOP), scalar VALU ----------------
__global__ void __launch_bounds__(TPB) node_encoder_kernel(
    const float* __restrict__ x,
    const float* __restrict__ ws, const float* __restrict__ bs,
    const float* __restrict__ wh, const float* __restrict__ bh,
    const float* __restrict__ we, const float* __restrict__ be,
    float* __restrict__ h_node) {
  const long n = (long)blockIdx.x * blockDim.x + threadIdx.x;
  if (n >= NNODES) return;
  const float xv = x[n];
  float h[DD], h2[DD];
#pragma unroll
  for (int d = 0; d < DD; ++d) h[d] = fmaxf(xv * ws[d] + bs[d], 0.0f);
  for (int L = 0; L < 2; ++L) {                 // shared hidden layer, twice
    for (int d = 0; d < DD; ++d) {
      float s = bh[d];
#pragma unroll
      for (int k = 0; k < DD; ++k) s += h[k] * wh[k * DD + d];
      h2[d] = fmaxf(s, 0.0f);
    }
#pragma unroll
    for (int d = 0; d < DD; ++d) h[d] = h2[d];
  }
  for (int d = 0; d < DD; ++d) {
    float s = be[d];
#pragma unroll
    for (int k = 0; k < DD; ++k) s += h[k] * we[k * DD + d];
    h_node[n * DD + d] = fmaxf(s, 0.0f);
  }
}

// ---------------- edge encoder: 3->32->32->32->32, WMMA, chained via LDS -----
__global__ void __launch_bounds__(TPB) edge_encoder_kernel(
    const float* __restrict__ ea,
    const float* __restrict__ Ws, const float* __restrict__ bs,
    const float* __restrict__ Wh, const float* __restrict__ bh,
    const float* __restrict__ We, const float* __restrict__ be,
    float* __restrict__ h_edge) {
  __shared__ __align__(16) float lds[WAVES * ROWS * RS];
  const int wave = threadIdx.x >> 5;
  const int lane = threadIdx.x & 31;
  float* tile = lds + wave * ROWS * RS;
  const long base = ((long)blockIdx.x * WAVES + wave) * ROWS;

  if (lane < ROWS) {
    const long e = base + lane;
    tile[lane * RS + 0] = ea[e * 3 + 0];
    tile[lane * RS + 1] = ea[e * 3 + 1];
    tile[lane * RS + 2] = ea[e * 3 + 2];
    tile[lane * RS + 3] = 0.0f;
  }
  __syncthreads();

  v8f acc[2];
  gemm_tile<3, 32, 2>(tile, Ws, bs, acc);
  __syncthreads();
  tile_to_lds<2>(tile, acc, true);
  __syncthreads();
  gemm_tile<32, 32, 2>(tile, Wh, bh, acc);
  __syncthreads();
  tile_to_lds<2>(tile, acc, true);
  __syncthreads();
  gemm_tile<32, 32, 2>(tile, Wh, bh, acc);
  __syncthreads();
  tile_to_lds<2>(tile, acc, true);
  __syncthreads();
  gemm_tile<32, 32, 2>(tile, We, be, acc);

  const int g = lane >> 4, col = lane & 15;
#pragma unroll
  for (int nt = 0; nt < 2; ++nt)
#pragma unroll
    for (int v = 0; v < 8; ++v)
      h_edge[(base + v + 8 * g) * DD + nt * 16 + col] = fmaxf(acc[nt][v], 0.0f);
}

// ---------------- message: relu([h_node[src]|h_edge] @ W64x32), scatter-add --
__global__ void __launch_bounds__(TPB) message_kernel(
    const float* __restrict__ h_node, const float* __restrict__ h_edge,
    const int* __restrict__ src, const int* __restrict__ dst,
    const float* __restrict__ W, float* __restrict__ accum) {
  __shared__ __align__(16) float lds[WAVES * ROWS * RS];
  const int wave = threadIdx.x >> 5;
  const int lane = threadIdx.x & 31;
  float* tile = lds + wave * ROWS * RS;
  const long base = ((long)blockIdx.x * WAVES + wave) * ROWS;

  {  // 2 lanes per row: half 0 = gathered node row, half 1 = edge row
    const int r = lane >> 1, half = lane & 1;
    const long e = base + r;
    const float* srow = half ? (h_edge + e * DD) : (h_node + (long)src[e] * DD);
    const float4* s4 = (const float4*)srow;
    float4* d4 = (float4*)(tile + r * RS + half * DD);
#pragma unroll
    for (int i = 0; i < 8; ++i) d4[i] = s4[i];
  }
  __syncthreads();

  v8f acc[2];
  gemm_tile<64, 32, 2>(tile, W, nullptr, acc);

  const int g = lane >> 4, col = lane & 15;
#pragma unroll
  for (int nt = 0; nt < 2; ++nt)
#pragma unroll
    for (int v = 0; v < 8; ++v) {
      const long e = base + v + 8 * g;
      const int dn = dst[e];
      atomicAdd(accum + (long)dn * DD + nt * 16 + col, fmaxf(acc[nt][v], 0.0f));
    }
}

// ---------------- edge update: relu([h_edge|h_n[src]|h_n[dst]] @ W96x32 + b) -
__global__ void __launch_bounds__(TPB) edge_update_kernel(
    float* __restrict__ h_edge, const float* __restrict__ h_node,
    const int* __restrict__ src, const int* __restrict__ dst,
    const float* __restrict__ W, const float* __restrict__ bias) {
  __shared__ __align__(16) float lds[WAVES * ROWS * RS];
  const int wave = threadIdx.x >> 5;
  const int lane = threadIdx.x & 31;
  float* tile = lds + wave * ROWS * RS;
  const long base = ((long)blockIdx.x * WAVES + wave) * ROWS;

  for (int s = lane; s < 48; s += 32) {   // 16 rows x 3 segments of 32 floats
    const int r = s / 3, part = s % 3;
    const long e = base + r;
    const float* srow;
    if (part == 0)      srow = h_edge + e * DD;
    else if (part == 1) srow = h_node + (long)src[e] * DD;
    else                srow = h_node + (long)dst[e] * DD;
    const float4* s4 = (const float4*)srow;
    float4* d4 = (float4*)(tile + r * RS + part * DD);
#pragma unroll
    for (int i = 0; i < 8; ++i) d4[i] = s4[i];
  }
  __syncthreads();

  v8f acc[2];
  gemm_tile<96, 32, 2>(tile, W, bias, acc);

  const int g = lane >> 4, col = lane & 15;   // in-place: rows already staged
#pragma unroll
  for (int nt = 0; nt < 2; ++nt)
#pragma unroll
    for (int v = 0; v < 8; ++v)
      h_edge[(base + v + 8 * g) * DD + nt * 16 + col] = fmaxf(acc[nt][v], 0.0f);
}

// ---------------- final edge layer: 96->3, no relu, out = edge_attr + res ----
__global__ void __launch_bounds__(TPB) edge_final_kernel(
    const float* __restrict__ h_edge, const float* __restrict__ h_node,
    const int* __restrict__ src, const int* __restrict__ dst,
    const float* __restrict__ W, const float* __restrict__ bias,
    const float* __restrict__ ea, float* __restrict__ out) {
  __shared__ __align__(16) float lds[WAVES * ROWS * RS];
  const int wave = threadIdx.x >> 5;
  const int lane = threadIdx.x & 31;
  float* tile = lds + wave * ROWS * RS;
  const long base = ((long)blockIdx.x * WAVES + wave) * ROWS;

  for (int s = lane; s < 48; s += 32) {
    const int r = s / 3, part = s % 3;
    const long e = base + r;
    const float* srow;
    if (part == 0)      srow = h_edge + e * DD;
    else if (part == 1) srow = h_node + (long)src[e] * DD;
    else                srow = h_node + (long)dst[e] * DD;
    const float4* s4 = (const float4*)srow;
    float4* d4 = (float4*)(tile + r * RS + part * DD);
#pragma unroll
    for (int i = 0; i < 8; ++i) d4[i] = s4[i];
  }
  __syncthreads();

  v8f acc[1];
  gemm_tile<96, 3, 1>(tile, W, bias, acc);

  const int g = lane >> 4, col = lane & 15;
  if (col < 3) {
#pragma unroll
    for (int v = 0; v < 8; ++v) {
      const long e = base + v + 8 * g;
      out[e * 3 + col] = ea[e * 3 + col] + acc[0][v];
    }
  }
}

// ---------------- small utility kernels --------------------------------------
__global__ void zero_kernel(float* p, long n) {
  const long i = (long)blockIdx.x * blockDim.x + threadIdx.x;
  if (i < n) p[i] = 0.0f;
}
__global__ void count_kernel(const int* __restrict__ dst, float* __restrict__ cnt) {
  const long i = (long)blockIdx.x * blockDim.x + threadIdx.x;
  if (i < NEDGES) atomicAdd(cnt + dst[i], 1.0f);
}
__global__ void invdeg_kernel(const float* __restrict__ cnt, float* __restrict__ inv) {
  const long i = (long)blockIdx.x * blockDim.x + threadIdx.x;
  if (i < NNODES) inv[i] = 1.0f / fmaxf(cnt[i], 1.0f);
}
__global__ void finalize_kernel(const float* __restrict__ accum,
                                const float* __restrict__ inv,
                                float* __restrict__ h_node) {
  const long i = (long)blockIdx.x * blockDim.x + threadIdx.x;
  if (i < (long)NNODES * DD) h_node[i] = accum[i] * inv[i >> 5];
}

extern "C" void kernel_launch(void* const* d_in, const int* in_sizes, int n_in,
                              void* d_out, int out_size, void* d_ws, size_t ws_size,
                              hipStream_t stream) {
  const float* x         = (const float*)d_in[0];
  const float* edge_attr = (const float*)d_in[1];
  const int*   eidx      = (const int*)d_in[2];
  const int*   src = eidx;
  const int*   dst = eidx + NEDGES;

  const float* w_enc_node_s = (const float*)d_in[3];
  const float* b_enc_node_s = (const float*)d_in[4];
  const float* w_enc_node_h = (const float*)d_in[5];
  const float* b_enc_node_h = (const float*)d_in[6];
  const float* w_enc_node_e = (const float*)d_in[7];
  const float* b_enc_node_e = (const float*)d_in[8];
  const float* w_enc_edge_s = (const float*)d_in[9];
  const float* b_enc_edge_s = (const float*)d_in[10];
  const float* w_enc_edge_h = (const float*)d_in[11];
  const float* b_enc_edge_h = (const float*)d_in[12];
  const float* w_enc_edge_e = (const float*)d_in[13];
  const float* b_enc_edge_e = (const float*)d_in[14];
  const float* w_node_s = (const float*)d_in[15];
  const float* w_node_h = (const float*)d_in[16];
  const float* w_node_e = (const float*)d_in[17];
  const float* w_edge_s = (const float*)d_in[18];
  const float* b_edge_s = (const float*)d_in[19];
  const float* w_edge_h = (const float*)d_in[20];
  const float* b_edge_h = (const float*)d_in[21];
  const float* w_edge_e = (const float*)d_in[22];
  const float* b_edge_e = (const float*)d_in[23];

  // workspace layout (floats)
  float* h_edge = (float*)d_ws;                      // E*32
  float* h_node = h_edge + (long)NEDGES * DD;        // N*32
  float* accum  = h_node + (long)NNODES * DD;        // N*32
  float* inv    = accum  + (long)NNODES * DD;        // N
  float* cnt    = inv    + NNODES;                   // N

  const int eblocks = NEDGES / EBLK;                 // 12500
  const int nblocks = (NNODES + TPB - 1) / TPB;
  const long nelems = (long)NNODES * DD;

  // encoders
  node_encoder_kernel<<<nblocks, TPB, 0, stream>>>(
      x, w_enc_node_s, b_enc_node_s, w_enc_node_h, b_enc_node_h,
      w_enc_node_e, b_enc_node_e, h_node);
  edge_encoder_kernel<<<eblocks, TPB, 0, stream>>>(
      edge_attr, w_enc_edge_s, b_enc_edge_s, w_enc_edge_h, b_enc_edge_h,
      w_enc_edge_e, b_enc_edge_e, h_edge);

  // in-degree reciprocals (invariant across blocks)
  zero_kernel<<<nblocks, TPB, 0, stream>>>(cnt, (long)NNODES);
  count_kernel<<<(NEDGES + TPB - 1) / TPB, TPB, 0, stream>>>(dst, cnt);
  invdeg_kernel<<<nblocks, TPB, 0, stream>>>(cnt, inv);

  const float* Wn[4] = {w_node_s, w_node_h, w_node_h, w_node_e};
  const float* We[3] = {w_edge_s, w_edge_h, w_edge_h};
  const float* Be[3] = {b_edge_s, b_edge_h, b_edge_h};

  for (int blk = 0; blk < 4; ++blk) {
    zero_kernel<<<(int)((nelems + TPB - 1) / TPB), TPB, 0, stream>>>(accum, nelems);
    message_kernel<<<eblocks, TPB, 0, stream>>>(h_node, h_edge, src, dst, Wn[blk], accum);
    finalize_kernel<<<(int)((nelems + TPB - 1) / TPB), TPB, 0, stream>>>(accum, inv, h_node);
    if (blk < 3) {
      edge_update_kernel<<<eblocks, TPB, 0, stream>>>(h_edge, h_node, src, dst,
                                                      We[blk], Be[blk]);
    } else {
      edge_final_kernel<<<eblocks, TPB, 0, stream>>>(h_edge, h_node, src, dst,
                                                     w_edge_e, b_edge_e, edge_attr,
                                                     (float*)d_out);
    }
  }
}